// DMoN_3882650435587
// MI455X (gfx1250) — compile-verified
//
#include <hip/hip_runtime.h>
#include <hip/hip_bf16.h>
#include <math.h>

typedef float v2f __attribute__((ext_vector_type(2)));
typedef float v8f __attribute__((ext_vector_type(8)));

#define NROWS 100000
#define NEDGE 3200000
#define DDIM  384
#define KCL   16
#define NTILES (NROWS/16)        // 6250, exact

// workspace layout (float offsets)
#define WS_MSUM 0                 // scalar: sum of edge weights (2m basis)
#define WS_T1   1                 // scalar: trace(graph_pooled) numerator
#define WS_V    16                // 16: A^T degrees
#define WS_CS   32                // 16: cluster sizes
#define WS_MMAT 64                // 16*384: unnormalized A^T F
#define WS_DEG  8192              // 100000: degrees
#define WS_TOTAL (WS_DEG + NROWS)

// output layout (float offsets, tuple flattened in return order)
#define OUT_FP   0                            // 16*384 features_pooled
#define OUT_AS   (KCL*DDIM)                   // 100000*16 assignments
#define OUT_SPEC (OUT_AS + NROWS*KCL)         // scalar
#define OUT_COLL (OUT_SPEC + 1)               // scalar

#define MACC_STRIDE 388   // 384 + 4 pad so lane halves hit disjoint LDS banks
#define ATILE_STRIDE 17   // 16 + 1 pad for transposed staging

__global__ void k_init(float* __restrict__ ws) {
    int i = blockIdx.x * blockDim.x + threadIdx.x;
    if (i < WS_TOTAL) ws[i] = 0.0f;
}

// Fused: logits = F@W+b (WMMA), softmax, assignments out, cluster_sizes,
// and unnormalized A^T F (WMMA) accumulated into ws M.
__global__ __launch_bounds__(256)
void k_assign(const float* __restrict__ feat, const float* __restrict__ Wg,
              const float* __restrict__ bg, float* __restrict__ assign_out,
              float* __restrict__ ws) {
    __shared__ float wlds[DDIM * KCL];              // 24.0 KB  W as [d][k]
    __shared__ float atile[8][16 * ATILE_STRIDE];   //  8.5 KB  per-wave staging [row][k]
    __shared__ float macc[KCL * MACC_STRIDE];       // 24.3 KB  workgroup M partial

    const int tid = threadIdx.x;
    for (int i = tid; i < DDIM * KCL; i += 256) wlds[i] = Wg[i];
    for (int i = tid; i < KCL * MACC_STRIDE; i += 256) macc[i] = 0.0f;
    __syncthreads();

    const int lane = tid & 31;
    const int wv   = tid >> 5;
    const int h    = lane >> 4;   // half-wave index
    const int lm   = lane & 15;

    float bj[8];
#pragma unroll
    for (int j = 0; j < 8; ++j) bj[j] = bg[j + 8 * h];

    float csacc[8] = {0.f, 0.f, 0.f, 0.f, 0.f, 0.f, 0.f, 0.f};
    float* myt = &atile[wv][0];

    const int wstride = gridDim.x * 8;
    for (int t = blockIdx.x * 8 + wv; t < NTILES; t += wstride) {
        const int row0 = t * 16;

        // ---- logits^T tile: C'(cluster,row) = W^T(16xD) x F^T(Dx16) ----
        v8f c = {};
#pragma unroll 4
        for (int s = 0; s < DDIM / 4; ++s) {
            const int d0 = 4 * s + 2 * h;
            v2f af;  // A = W^T : (M=cluster=lm, K=d0..d0+1)
            af.x = wlds[d0 * KCL + lm];
            af.y = wlds[(d0 + 1) * KCL + lm];
            // B = F^T : (K=d0..d0+1, N=row=lm) -> contiguous float2 in a row
            v2f bf = *(const v2f*)(feat + (row0 + lm) * DDIM + d0);
            c = __builtin_amdgcn_wmma_f32_16x16x4_f32(
                    false, af, false, bf, (short)0, c, false, false);
        }

        // ---- bias + softmax over 16 clusters (8 here + 8 in other half) ----
        float e[8];
        float mx = -3.0e38f;
#pragma unroll
        for (int j = 0; j < 8; ++j) { e[j] = c[j] + bj[j]; mx = fmaxf(mx, e[j]); }
        mx = fmaxf(mx, __shfl_xor(mx, 16));
        float s8 = 0.0f;
#pragma unroll
        for (int j = 0; j < 8; ++j) { e[j] = __expf(e[j] - mx); s8 += e[j]; }
        s8 += __shfl_xor(s8, 16);
        const float inv = 1.0f / s8;
#pragma unroll
        for (int j = 0; j < 8; ++j) { e[j] *= inv; csacc[j] += e[j]; }

        // ---- stage a-tile transposed to [row][cluster] in LDS ----
#pragma unroll
        for (int j = 0; j < 8; ++j) myt[lm * ATILE_STRIDE + j + 8 * h] = e[j];

        // ---- coalesced assignments store: 8 consecutive floats per lane ----
        {
            const int fi = lane * 8;  // wave covers all 256 tile elements
            const int base = (fi >> 4) * ATILE_STRIDE + (fi & 15);
            float4 p0, p1;
            p0.x = myt[base + 0]; p0.y = myt[base + 1];
            p0.z = myt[base + 2]; p0.w = myt[base + 3];
            p1.x = myt[base + 4]; p1.y = myt[base + 5];
            p1.z = myt[base + 6]; p1.w = myt[base + 7];
            float4* dst = (float4*)(assign_out + row0 * KCL + fi);
            dst[0] = p0; dst[1] = p1;
        }

        // ---- M partial: a^T(16x16) x f_tile(16x384) via WMMA ----
        v2f am[4];  // A = a^T : (M=cluster=lm, K=row_local), reused for all 24 d-tiles
#pragma unroll
        for (int s = 0; s < 4; ++s) {
            am[s].x = myt[(4 * s + 2 * h) * ATILE_STRIDE + lm];
            am[s].y = myt[(4 * s + 2 * h + 1) * ATILE_STRIDE + lm];
        }
#pragma unroll 2
        for (int dt = 0; dt < DDIM / 16; ++dt) {
            const int db = dt * 16;
            v8f mc = {};
#pragma unroll
            for (int s = 0; s < 4; ++s) {
                v2f bf;  // B = f_tile : (K=row_local, N=d) -> coalesced per half
                bf.x = feat[(row0 + 4 * s + 2 * h) * DDIM + db + lm];
                bf.y = feat[(row0 + 4 * s + 2 * h + 1) * DDIM + db + lm];
                mc = __builtin_amdgcn_wmma_f32_16x16x4_f32(
                        false, am[s], false, bf, (short)0, mc, false, false);
            }
#pragma unroll
            for (int j = 0; j < 8; ++j)
                atomicAdd(&macc[(j + 8 * h) * MACC_STRIDE + db + lm], mc[j]);
        }
    }

    // ---- cluster_sizes flush (reduce across 16 lanes within each half) ----
#pragma unroll
    for (int j = 0; j < 8; ++j) {
        float v = csacc[j];
        v += __shfl_xor(v, 1); v += __shfl_xor(v, 2);
        v += __shfl_xor(v, 4); v += __shfl_xor(v, 8);
        if (lm == 0) atomicAdd(&ws[WS_CS + j + 8 * h], v);
    }

    __syncthreads();
    // ---- one global M flush per workgroup ----
    for (int i = tid; i < KCL * DDIM; i += 256)
        atomicAdd(&ws[WS_MMAT + i], macc[(i / DDIM) * MACC_STRIDE + (i % DDIM)]);
}

__device__ __forceinline__ void block_reduce_atomic(float acc, float* target) {
    acc += __shfl_xor(acc, 1);  acc += __shfl_xor(acc, 2);
    acc += __shfl_xor(acc, 4);  acc += __shfl_xor(acc, 8);
    acc += __shfl_xor(acc, 16);
    __shared__ float red[8];
    if ((threadIdx.x & 31) == 0) red[threadIdx.x >> 5] = acc;
    __syncthreads();
    if (threadIdx.x == 0) {
        float s = 0.0f;
#pragma unroll
        for (int j = 0; j < 8; ++j) s += red[j];
        atomicAdd(target, s);
    }
}

// degrees[dst] += w ; m = sum(w)
__global__ __launch_bounds__(256)
void k_deg(const int* __restrict__ edst, const float* __restrict__ ew,
           float* __restrict__ ws) {
    float acc = 0.0f;
    const int stride = gridDim.x * blockDim.x;
    for (int i = blockIdx.x * blockDim.x + threadIdx.x; i < NEDGE; i += stride) {
        const float w = ew[i];
        acc += w;
        atomicAdd(&ws[WS_DEG + edst[i]], w);
    }
    block_reduce_atomic(acc, &ws[WS_MSUM]);
}

// t1 = sum_e w_e * dot(a[dst], a[src])  (assignments resident in L2)
__global__ __launch_bounds__(256)
void k_trace(const int* __restrict__ esrc, const int* __restrict__ edst,
             const float* __restrict__ ew, const float* __restrict__ assign,
             float* __restrict__ ws) {
    float acc = 0.0f;
    const int stride = gridDim.x * blockDim.x;
    for (int i = blockIdx.x * blockDim.x + threadIdx.x; i < NEDGE; i += stride) {
        const float4* a = (const float4*)(assign + (size_t)esrc[i] * KCL);
        const float4* b = (const float4*)(assign + (size_t)edst[i] * KCL);
        float d = 0.0f;
#pragma unroll
        for (int q = 0; q < 4; ++q) {
            float4 x = a[q], y = b[q];
            d += x.x * y.x + x.y * y.y + x.z * y.z + x.w * y.w;
        }
        acc += ew[i] * d;
    }
    block_reduce_atomic(acc, &ws[WS_T1]);
}

// v[k] = sum_i a[i,k] * degrees[i]; warp handles 2 rows x 16 k per step
__global__ __launch_bounds__(256)
void k_vvec(const float* __restrict__ assign, const float* __restrict__ deg,
            float* __restrict__ ws) {
    const int lane = threadIdx.x & 31, h = lane >> 4, lm = lane & 15;
    const int gw = (blockIdx.x * blockDim.x + threadIdx.x) >> 5;
    const int nw = (gridDim.x * blockDim.x) >> 5;
    float acc = 0.0f;
    for (int p = gw; p < NROWS / 2; p += nw) {
        const int r = 2 * p + h;
        acc += assign[r * KCL + lm] * deg[r];
    }
    acc += __shfl_xor(acc, 16);
    __shared__ float vred[16];
    if (threadIdx.x < 16) vred[threadIdx.x] = 0.0f;
    __syncthreads();
    if (lane < 16) atomicAdd(&vred[lm], acc);
    __syncthreads();
    if (threadIdx.x < 16) atomicAdd(&ws[WS_V + threadIdx.x], vred[threadIdx.x]);
}

// losses + features_pooled = selu(M / cluster_sizes)
__global__ __launch_bounds__(256)
void k_final(const float* __restrict__ ws, float* __restrict__ out) {
    const int tid = threadIdx.x;
    if (tid == 0) {
        const float m = ws[WS_MSUM];
        const float t1 = ws[WS_T1];
        float vv = 0.0f, cn = 0.0f;
#pragma unroll
        for (int k = 0; k < KCL; ++k) {
            const float v = ws[WS_V + k];  vv += v * v;
            const float c = ws[WS_CS + k]; cn += c * c;
        }
        const float inv2m = 1.0f / (2.0f * m);
        out[OUT_SPEC] = -(t1 - vv * inv2m) * inv2m;
        out[OUT_COLL] = 0.1f * (sqrtf(cn) / (float)NROWS * 4.0f - 1.0f);
    }
    const float sc = 1.0507009873554805f, al = 1.6732632423543772f;
    for (int i = tid; i < KCL * DDIM; i += 256) {
        const float x = ws[WS_MMAT + i] / ws[WS_CS + i / DDIM];
        out[OUT_FP + i] = x > 0.0f ? sc * x : sc * al * (__expf(x) - 1.0f);
    }
}

extern "C" void kernel_launch(void* const* d_in, const int* in_sizes, int n_in,
                              void* d_out, int out_size, void* d_ws, size_t ws_size,
                              hipStream_t stream) {
    (void)in_sizes; (void)n_in; (void)out_size; (void)ws_size;
    const float* feat = (const float*)d_in[0];
    const int*   esrc = (const int*)d_in[1];
    const int*   edst = (const int*)d_in[2];
    const float* ew   = (const float*)d_in[3];
    const float* Wg   = (const float*)d_in[4];
    const float* bg   = (const float*)d_in[5];
    float* out = (float*)d_out;
    float* ws  = (float*)d_ws;

    k_init<<<(WS_TOTAL + 255) / 256, 256, 0, stream>>>(ws);
    k_assign<<<320, 256, 0, stream>>>(feat, Wg, bg, out + OUT_AS, ws);
    k_deg<<<2048, 256, 0, stream>>>(edst, ew, ws);
    k_trace<<<2048, 256, 0, stream>>>(esrc, edst, ew, out + OUT_AS, ws);
    k_vvec<<<256, 256, 0, stream>>>(out + OUT_AS, ws + WS_DEG, ws);
    k_final<<<1, 256, 0, stream>>>(ws, out);
}